// OTNoiseGate_85126251807319
// MI455X (gfx1250) — compile-verified
//
#include <hip/hip_runtime.h>

typedef __attribute__((ext_vector_type(2))) float v2f;
typedef __attribute__((ext_vector_type(8))) float v8f;

#define NEG_INV_EPS_LN2 (-28.853900817779268f)   // -1/(0.05*ln2)
#define NEG_EPS_LN2     (-0.034657359027997264f) // -0.05*ln2
#define LOG2E           (1.4426950408889634f)

// ---- per-row squared-norm: one wave32 per row of 512 floats ----
__global__ void rowsq_kernel(const float* __restrict__ src,
                             float* __restrict__ dst, int nrows) {
  int row  = blockIdx.x * (blockDim.x >> 5) + (threadIdx.x >> 5);
  int lane = threadIdx.x & 31;
  if (row >= nrows) return;
  const float4* p = (const float4*)(src + (size_t)row * 512);
  float acc = 0.f;
#pragma unroll
  for (int i = 0; i < 4; ++i) {
    float4 v = p[lane + 32 * i];
    acc += v.x * v.x + v.y * v.y + v.z * v.z + v.w * v.w;
  }
#pragma unroll
  for (int off = 16; off > 0; off >>= 1) acc += __shfl_xor(acc, off, 32);
  if (lane == 0) dst[row] = acc;
}

__device__ __forceinline__ v8f wmma4(v2f a, v2f b, v8f c) {
  // V_WMMA_F32_16X16X4_F32 : D = A(16x4) x B(4x16) + C(16x16), f32
  return __builtin_amdgcn_wmma_f32_16x16x4_f32(false, a, false, b, (short)0, c,
                                               false, false);
}

__launch_bounds__(64)
__global__ void otgate_kernel(const float* __restrict__ tok,
                              const float* __restrict__ prot,
                              const int* __restrict__ fiPtr,
                              const float* __restrict__ s2g,
                              const float* __restrict__ p2g,
                              float* __restrict__ out) {
  constexpr int KC = 64;
  __shared__ float ldsProto[64 * (KC + 4)]; // [ee*32+p][kk], stride 68 (pad)
  __shared__ float ldsCost[2][32 * 72];     // per wave: [mat][s*36 + j]
  __shared__ float ldsP2[256];

  const int tid  = threadIdx.x;
  const int wid  = tid >> 5;
  const int lane = tid & 31;
  const int lm   = lane & 15;
  const int lh   = lane >> 4;
  const int fi   = *fiPtr;
  const int b0   = blockIdx.x * 32 + wid * 16;

  for (int i = tid; i < 256; i += 64) ldsP2[i] = p2g[i];

  const int rowb = b0 + lm; // A-fragment row owned by this lane
  const float* aAp = tok + ((size_t)rowb * 8 + fi) * 512 + 2 * lh; // anchor

  const float* bp[2][2];
#pragma unroll
  for (int ee = 0; ee < 2; ++ee)
#pragma unroll
    for (int nt = 0; nt < 2; ++nt)
      bp[ee][nt] = &ldsProto[(ee * 32 + nt * 16 + lm) * 68 + 2 * lh];

  for (int ep = 0; ep < 4; ++ep) {
    const int e0 = ep * 2;
    const float* a0p = tok + ((size_t)rowb * 8 + e0) * 512 + 2 * lh;
    const float* a1p = a0p + 512;

    v8f acc[2][2][2]; // [expert-local][s][ntile]
    const v8f vzero = {0.f, 0.f, 0.f, 0.f, 0.f, 0.f, 0.f, 0.f};
#pragma unroll
    for (int x = 0; x < 2; ++x)
#pragma unroll
      for (int y = 0; y < 2; ++y)
#pragma unroll
        for (int z = 0; z < 2; ++z) acc[x][y][z] = vzero;

    for (int kc = 0; kc < 8; ++kc) {
      __syncthreads();
      // stage prototypes[e0..e0+1][:, kc*64 .. +63] into padded LDS
      for (int i = tid; i < 64 * 16; i += 64) {
        int row = i >> 4, c4 = i & 15;
        int ee = row >> 5, p = row & 31;
        float4 v = *(const float4*)(prot +
                    (((size_t)(e0 + ee) * 32 + p) * 512 + kc * KC + c4 * 4));
        *(float4*)(&ldsProto[row * 68 + c4 * 4]) = v;
      }
      __syncthreads();
#pragma unroll
      for (int kk = 0; kk < KC; kk += 4) {
        const int k0 = kc * KC + kk;
        v2f a0 = *(const v2f*)(a0p + k0);
        v2f a1 = *(const v2f*)(a1p + k0);
        v2f aA = *(const v2f*)(aAp + k0);
        v2f b00 = *(const v2f*)(bp[0][0] + kk);
        v2f b01 = *(const v2f*)(bp[0][1] + kk);
        v2f b10 = *(const v2f*)(bp[1][0] + kk);
        v2f b11 = *(const v2f*)(bp[1][1] + kk);
        acc[0][0][0] = wmma4(a0, b00, acc[0][0][0]);
        acc[0][0][1] = wmma4(a0, b01, acc[0][0][1]);
        acc[0][1][0] = wmma4(aA, b00, acc[0][1][0]);
        acc[0][1][1] = wmma4(aA, b01, acc[0][1][1]);
        acc[1][0][0] = wmma4(a1, b10, acc[1][0][0]);
        acc[1][0][1] = wmma4(a1, b11, acc[1][0][1]);
        acc[1][1][0] = wmma4(aA, b10, acc[1][1][0]);
        acc[1][1][1] = wmma4(aA, b11, acc[1][1][1]);
      }
    }

    // C tiles -> per-wave LDS (transpose fragment layout to per-matrix rows)
    float* cw = ldsCost[wid];
#pragma unroll
    for (int tee = 0; tee < 2; ++tee)
#pragma unroll
      for (int ts = 0; ts < 2; ++ts)
#pragma unroll
        for (int tnt = 0; tnt < 2; ++tnt)
#pragma unroll
          for (int r = 0; r < 8; ++r) {
            int bl = r + 8 * lh;
            cw[(bl * 2 + tee) * 72 + ts * 36 + tnt * 16 + lm] =
                acc[tee][ts][tnt][r];
          }

    // ---- Sinkhorn: one lane per (b_local, expert) 2x32 matrix, log2 domain
    {
      const int bl = lane >> 1;
      const int ee = lane & 1;
      const int eg = e0 + ee;
      const int b  = b0 + bl;
      const float s2t = s2g[(size_t)b * 8 + eg];
      const float s2a = s2g[(size_t)b * 8 + fi];
      const float* mbase = cw + (bl * 2 + ee) * 72;

      float k2[2][32];
#pragma unroll
      for (int s = 0; s < 2; ++s) {
        const float sv = s ? s2a : s2t;
#pragma unroll
        for (int j = 0; j < 32; ++j) {
          float cr = mbase[s * 36 + j];
          float c  = fmaxf(sv - 2.0f * cr + ldsP2[eg * 32 + j], 0.0f);
          k2[s][j] = c * NEG_INV_EPS_LN2; // log2(K) = -cost/(eps*ln2)
        }
      }

      float vv[32];
#pragma unroll
      for (int j = 0; j < 32; ++j) vv[j] = 0.0f;
      float u0 = 0.0f, u1 = 0.0f;

      for (int it = 0; it < 50; ++it) {
#pragma unroll
        for (int i = 0; i < 2; ++i) {
          float aj[32];
          float m = -3.0e38f;
#pragma unroll
          for (int j = 0; j < 32; ++j) {
            aj[j] = k2[i][j] + vv[j];
            m = fmaxf(m, aj[j]);
          }
          float sum = 0.0f;
#pragma unroll
          for (int j = 0; j < 32; ++j)
            sum += __builtin_amdgcn_exp2f(aj[j] - m);
          float u = -1.0f - (m + __builtin_amdgcn_logf(sum)); // log2(1/2) = -1
          if (i == 0) u0 = u; else u1 = u;
        }
#pragma unroll
        for (int j = 0; j < 32; ++j) {
          float t0 = k2[0][j] + u0;
          float t1 = k2[1][j] + u1;
          float hi = fmaxf(t0, t1);
          float lo = fminf(t0, t1);
          // log2(1/32) = -5 ; LSE2 over the 2 rows
          vv[j] = -5.0f - (hi + __builtin_amdgcn_logf(
                                    1.0f + __builtin_amdgcn_exp2f(lo - hi)));
        }
      }

      float a2 = 0.0f;
#pragma unroll
      for (int j = 0; j < 32; ++j) {
        a2 += k2[0][j] * __builtin_amdgcn_exp2f(k2[0][j] + u0 + vv[j]);
        a2 += k2[1][j] * __builtin_amdgcn_exp2f(k2[1][j] + u1 + vv[j]);
      }
      float ot = a2 * NEG_EPS_LN2; // sum(transport * cost)
      float valid =
          1.0f / (1.0f + __builtin_amdgcn_exp2f(6.0f * (ot - 0.25f) * LOG2E));
      if (eg == fi) { ot = 0.0f; valid = 1.0f; }
      out[(size_t)b * 8 + eg]         = valid;
      out[65536 + (size_t)b * 8 + eg] = ot;
    }
  }
}

extern "C" void kernel_launch(void* const* d_in, const int* in_sizes, int n_in,
                              void* d_out, int out_size, void* d_ws,
                              size_t ws_size, hipStream_t stream) {
  (void)in_sizes; (void)n_in; (void)out_size; (void)ws_size;
  const float* tok  = (const float*)d_in[0];  // (8192,8,512) f32
  const float* prot = (const float*)d_in[1];  // (8,32,512) f32
  const int*   fi   = (const int*)d_in[2];    // scalar
  float* out = (float*)d_out;                 // 65536 validity + 65536 ot
  float* s2  = (float*)d_ws;                  // 65536 floats
  float* p2  = s2 + 65536;                    // 256 floats

  rowsq_kernel<<<8192, 256, 0, stream>>>(tok, s2, 65536);
  rowsq_kernel<<<32, 256, 0, stream>>>(prot, p2, 256);
  otgate_kernel<<<256, 64, 0, stream>>>(tok, prot, fi, s2, p2, out);
}